// GraphSAGE_59880434041043
// MI455X (gfx1250) — compile-verified
//
#include <hip/hip_runtime.h>
#include <hip/hip_bf16.h>

typedef __attribute__((ext_vector_type(2))) float v2f;
typedef __attribute__((ext_vector_type(8))) float v8f;

#define BN_EPS 1e-5f

// ---------------------------------------------------------------------------
// zero a float buffer (grid-stride)
// ---------------------------------------------------------------------------
__global__ void zero_f32(float* __restrict__ p, size_t n) {
    size_t i = (size_t)blockIdx.x * blockDim.x + threadIdx.x;
    size_t stride = (size_t)gridDim.x * blockDim.x;
    for (; i < n; i += stride) p[i] = 0.0f;
}

// ---------------------------------------------------------------------------
// degree count: one thread per edge, atomic f32 add into deg[dst]
// ---------------------------------------------------------------------------
__global__ void deg_count(const int* __restrict__ dst, float* __restrict__ deg, int E) {
    int e = blockIdx.x * blockDim.x + threadIdx.x;
    if (e < E) atomicAdd(&deg[dst[e]], 1.0f);
}

// deg -> 1/max(deg,1) in place
__global__ void deg_inv(float* __restrict__ deg, int N) {
    int i = blockIdx.x * blockDim.x + threadIdx.x;
    if (i < N) deg[i] = 1.0f / fmaxf(deg[i], 1.0f);
}

// ---------------------------------------------------------------------------
// edge scatter: one wave (32 lanes) per edge; each lane handles 4 features.
// float4 gather of h[src] (coalesced 512B per wave), 4 f32 atomics to agg[dst].
// ---------------------------------------------------------------------------
__global__ void scatter_edges(const float* __restrict__ h,
                              const int* __restrict__ src,
                              const int* __restrict__ dst,
                              float* __restrict__ agg, int E) {
    int wid  = (blockIdx.x * blockDim.x + threadIdx.x) >> 5;   // edge id
    int lane = threadIdx.x & 31;
    if (wid >= E) return;
    int s = src[wid];
    int d = dst[wid];
    const float4 v = *reinterpret_cast<const float4*>(h + (size_t)s * 128 + lane * 4);
    float* ap = agg + (size_t)d * 128 + lane * 4;
    atomicAdd(ap + 0, v.x);
    atomicAdd(ap + 1, v.y);
    atomicAdd(ap + 2, v.z);
    atomicAdd(ap + 3, v.w);
}

// ---------------------------------------------------------------------------
// SAGE GEMM: out[16 rows x NCOLS] per wave, fp32 WMMA 16x16x4.
//   out = (agg * dinv) @ Wl + h @ Wr + bias
// Weight matrix staged into LDS via gfx1250 async loads
// (global_load_async_to_lds_b128 + s_wait_asynccnt), B-fragments from DS.
// A layout (16x4 f32, ISA 7.12.2): lanes 0-15 hold row=lane, K={k,k+1};
//   lanes 16-31 hold row=lane-16, K={k+2,k+3}. B mirrored (col = lane&15).
// C/D: VGPR r: lanes 0-15 -> M=r, lanes 16-31 -> M=8+r; N = lane&15.
// ---------------------------------------------------------------------------
template <int NCOLS>
__global__ void sage_gemm(const float* __restrict__ agg,
                          const float* __restrict__ hin,
                          const float* __restrict__ dinv,
                          const float* __restrict__ Wl,
                          const float* __restrict__ Wr,
                          const float* __restrict__ bias,
                          float* __restrict__ out,
                          int nrows) {
    constexpr int NT = (NCOLS + 15) / 16;          // 16-wide column tiles
    __shared__ float smem[128 * NCOLS];            // one weight matrix (row-major)

    const int lane = threadIdx.x & 31;
    const int wave = threadIdx.x >> 5;
    const int waves_per_block = blockDim.x >> 5;
    const int base = (blockIdx.x * waves_per_block + wave) * 16;
    const bool active = (base < nrows);            // wave-uniform

    const int nl   = lane & 15;                    // col-in-tile (B/C) == row-in-tile (A)
    const int koff = (lane < 16) ? 0 : 2;
    const int row  = active ? (base + nl) : nl;    // clamped: inactive waves read row 0..15
    const float dscale = dinv[row];

    const v8f vzero = {0.f, 0.f, 0.f, 0.f, 0.f, 0.f, 0.f, 0.f};
    v8f acc[NT];
#pragma unroll
    for (int j = 0; j < NT; ++j) acc[j] = vzero;

#pragma unroll 1
    for (int m = 0; m < 2; ++m) {
        const float* __restrict__ Ap = m ? hin : agg;
        const float* __restrict__ Wp = m ? Wr : Wl;
        const float ascale = m ? 1.0f : dscale;

        // ---- stage weight matrix into LDS with async-to-LDS loads ----
        {
            const int chunks = (128 * NCOLS) / 4;  // 16-byte chunks
            for (int i = threadIdx.x; i < chunks; i += blockDim.x) {
                unsigned lds_addr = (unsigned)(i * 16);  // smem is at LDS offset 0
                unsigned goff     = (unsigned)(i * 16);
                asm volatile("global_load_async_to_lds_b128 %0, %1, %2"
                             :: "v"(lds_addr), "v"(goff), "s"(Wp)
                             : "memory");
            }
            asm volatile("s_wait_asynccnt 0x0" ::: "memory");
            __syncthreads();
        }

        // ---- K loop: fp32 WMMA 16x16x4, K = 128 ----
        for (int k = 0; k < 128; k += 4) {
            const float* ap = Ap + (size_t)row * 128 + (k + koff);
            v2f a;
            a.x = ap[0] * ascale;
            a.y = ap[1] * ascale;
#pragma unroll
            for (int j = 0; j < NT; ++j) {
                const int col   = j * 16 + nl;
                const bool okc  = (col < NCOLS);   // folds away except last partial tile
                const int cc    = okc ? col : 0;
                float bx = smem[(k + koff) * NCOLS + cc];
                float by = smem[(k + koff + 1) * NCOLS + cc];
                v2f b;
                b.x = okc ? bx : 0.f;
                b.y = okc ? by : 0.f;
                acc[j] = __builtin_amdgcn_wmma_f32_16x16x4_f32(
                    false, a, false, b, (short)0, acc[j], false, false);
            }
        }
        __syncthreads();  // done reading smem before next stage overwrites
    }

    const int rbase = base + ((lane >= 16) ? 8 : 0);
#pragma unroll
    for (int j = 0; j < NT; ++j) {
        const int col  = j * 16 + nl;
        const bool okc = (col < NCOLS);
        if (active && okc) {
            float bv = bias[col];
#pragma unroll
            for (int r = 0; r < 8; ++r) {
                out[(size_t)(rbase + r) * NCOLS + col] = acc[j][r] + bv;
            }
        }
    }
}

// ---------------------------------------------------------------------------
// BN column stats: blockDim=128 (thread <-> column), 256 rows per block,
// local accumulation then 2 atomics per column per block.
// stats[0..127]=sum, [128..255]=sumsq, [256..383]=scale, [384..511]=shift
// ---------------------------------------------------------------------------
__global__ void bn_stats(const float* __restrict__ h, float* __restrict__ stats, int nrows) {
    const int c  = threadIdx.x;  // 0..127
    const int r0 = blockIdx.x * 256;
    const int r1 = min(r0 + 256, nrows);
    float s = 0.f, q = 0.f;
    for (int r = r0; r < r1; ++r) {
        float v = h[(size_t)r * 128 + c];
        s += v;
        q += v * v;
    }
    atomicAdd(&stats[c], s);
    atomicAdd(&stats[128 + c], q);
}

__global__ void bn_finalize(float* __restrict__ stats,
                            const float* __restrict__ gamma,
                            const float* __restrict__ beta, int nrows) {
    int c = threadIdx.x;  // 128 threads
    float inv_n = 1.0f / (float)nrows;
    float mu  = stats[c] * inv_n;
    float var = stats[128 + c] * inv_n - mu * mu;
    float a = gamma[c] * rsqrtf(var + BN_EPS);
    stats[256 + c] = a;
    stats[384 + c] = beta[c] - mu * a;
}

__global__ void bn_apply_relu(float* __restrict__ h, const float* __restrict__ stats, size_t total) {
    size_t i = (size_t)blockIdx.x * blockDim.x + threadIdx.x;
    if (i >= total) return;
    int c = (int)(i & 127);
    float y = h[i] * stats[256 + c] + stats[384 + c];
    h[i] = fmaxf(y, 0.0f);
}

// ---------------------------------------------------------------------------
// log_softmax over 40 logits, thread per row, in place.
// ---------------------------------------------------------------------------
__global__ void log_softmax40(float* __restrict__ io, int nrows) {
    int r = blockIdx.x * blockDim.x + threadIdx.x;
    if (r >= nrows) return;
    float* p = io + (size_t)r * 40;
    float v[40];
    float mx = -3.402823466e38f;
#pragma unroll
    for (int j = 0; j < 40; ++j) {
        v[j] = p[j];
        mx = fmaxf(mx, v[j]);
    }
    float s = 0.f;
#pragma unroll
    for (int j = 0; j < 40; ++j) s += expf(v[j] - mx);
    float ls = logf(s) + mx;
#pragma unroll
    for (int j = 0; j < 40; ++j) p[j] = v[j] - ls;
}

// ---------------------------------------------------------------------------
// launch
// inputs: 0:x 1:edge_index 2-5:Wl0..3 6-9:bl0..3 10-13:Wr0..3 14-16:gamma 17-19:beta
// ---------------------------------------------------------------------------
extern "C" void kernel_launch(void* const* d_in, const int* in_sizes, int n_in,
                              void* d_out, int out_size, void* d_ws, size_t ws_size,
                              hipStream_t stream) {
    const float* x   = (const float*)d_in[0];
    const int*   ei  = (const int*)d_in[1];
    const float* Wl[4] = {(const float*)d_in[2], (const float*)d_in[3],
                          (const float*)d_in[4], (const float*)d_in[5]};
    const float* bl[4] = {(const float*)d_in[6], (const float*)d_in[7],
                          (const float*)d_in[8], (const float*)d_in[9]};
    const float* Wr[4] = {(const float*)d_in[10], (const float*)d_in[11],
                          (const float*)d_in[12], (const float*)d_in[13]};
    const float* gam[3] = {(const float*)d_in[14], (const float*)d_in[15], (const float*)d_in[16]};
    const float* bet[3] = {(const float*)d_in[17], (const float*)d_in[18], (const float*)d_in[19]};

    const int N = in_sizes[0] / 128;
    const int E = in_sizes[1] / 2;
    const int* src = ei;
    const int* dst = ei + E;

    float* ws = (float*)d_ws;
    float* deg   = ws;                                    // N floats
    float* stats = ws + (((size_t)N + 63) & ~(size_t)63); // 512 floats
    float* agg   = stats + 512;                           // N*128
    float* hA    = agg + (size_t)N * 128;                 // N*128
    float* hB    = hA + (size_t)N * 128;                  // N*128

    const size_t feat_total = (size_t)N * 128;
    const int ZB = 2048, ZT = 256;

    // degrees (shared by all layers)
    zero_f32<<<ZB, ZT, 0, stream>>>(deg, (size_t)N);
    deg_count<<<(E + 255) / 256, 256, 0, stream>>>(dst, deg, E);
    deg_inv<<<(N + 255) / 256, 256, 0, stream>>>(deg, N);

    const float* h_in = x;
    float* h_out[3] = {hA, hB, hA};

    // scatter grid: one wave per edge
    const int scat_blocks = (int)(((size_t)E * 32 + 255) / 256);
    const int gemm_blocks = (N + 127) / 128;  // 8 waves/block * 16 rows

    for (int L = 0; L < 3; ++L) {
        zero_f32<<<ZB, ZT, 0, stream>>>(agg, feat_total);
        scatter_edges<<<scat_blocks, 256, 0, stream>>>(h_in, src, dst, agg, E);
        sage_gemm<128><<<gemm_blocks, 256, 0, stream>>>(agg, h_in, deg, Wl[L], Wr[L],
                                                        bl[L], h_out[L], N);
        zero_f32<<<1, 256, 0, stream>>>(stats, (size_t)256);  // sum+sumsq
        bn_stats<<<(N + 255) / 256, 128, 0, stream>>>(h_out[L], stats, N);
        bn_finalize<<<1, 128, 0, stream>>>(stats, gam[L], bet[L], N);
        bn_apply_relu<<<(int)((feat_total + 255) / 256), 256, 0, stream>>>(h_out[L], stats, feat_total);
        h_in = h_out[L];
    }

    // final conv -> logits (40 cols) -> log_softmax
    float* logits = (float*)d_out;
    zero_f32<<<ZB, ZT, 0, stream>>>(agg, feat_total);
    scatter_edges<<<scat_blocks, 256, 0, stream>>>(h_in, src, dst, agg, E);
    sage_gemm<40><<<gemm_blocks, 256, 0, stream>>>(agg, h_in, deg, Wl[3], Wr[3],
                                                   bl[3], logits, N);
    log_softmax40<<<(N + 255) / 256, 256, 0, stream>>>(logits, N);
}